// SessionGraph_60601988546713
// MI455X (gfx1250) — compile-verified
//
#include <hip/hip_runtime.h>

#define EPSF 1e-8f

typedef float v2f __attribute__((ext_vector_type(2)));
typedef float v8f __attribute__((ext_vector_type(8)));

// C[M,N] = A[M,K] @ W[N,K]^T + bias[N], fp32 WMMA 16x16x4.
// Block = 256 threads = 8 waves; each wave owns a 16x16 tile; block covers 32(M) x 64(N).
__global__ __launch_bounds__(256) void gemm_nt_wmma(
    const float* __restrict__ A, const float* __restrict__ W,
    const float* __restrict__ bias, float* __restrict__ C,
    int M, int N, int K)
{
  const int lane = threadIdx.x & 31;
  const int wave = threadIdx.x >> 5;
  const int tm = blockIdx.x * 32 + (wave >> 2) * 16;
  const int tn = blockIdx.y * 64 + (wave & 3) * 16;
  const int l15 = lane & 15;
  const int koff = (lane >> 4) << 1;              // lanes 16..31 hold K+2,K+3
  const float* ap = A + (size_t)(tm + l15) * K + koff;
  const float* wp = W + (size_t)(tn + l15) * K + koff;
  v8f acc = {};
  for (int k = 0; k < K; k += 4) {
    v2f av = *(const v2f*)(ap + k);               // A frag: {A[m,k+off], A[m,k+off+1]}
    v2f bv = *(const v2f*)(wp + k);               // B frag: {W[n,k+off], W[n,k+off+1]} = B[k,n]
    acc = __builtin_amdgcn_wmma_f32_16x16x4_f32(false, av, false, bv,
                                                (short)0, acc, false, false);
  }
  const int n = tn + l15;
  const float bb = bias[n];
  const int mbase = tm + ((lane >> 4) << 3);      // lanes>=16 hold rows 8..15
  float* cp = C + (size_t)mbase * N + n;
#pragma unroll
  for (int r = 0; r < 8; ++r) cp[(size_t)r * N] = acc[r] + bb;
}

// One block per batch element. Computes scrambled-view scores, softmax->alpha,
// z_h weighted sums, a = W2-contraction + LayerNorm. Writes out[:, 0:256] and alpha.
__global__ __launch_bounds__(256) void attn_fused(
    const float* __restrict__ q0g, const float* __restrict__ hidden,
    const int* __restrict__ mask,
    const float* __restrict__ W1, const float* __restrict__ b1,
    const float* __restrict__ W2, const float* __restrict__ b2,
    const float* __restrict__ ln_g, const float* __restrict__ ln_b,
    float* __restrict__ out, float* __restrict__ alpha_out, int L)
{
  const int b = blockIdx.x;
  const int t = threadIdx.x;
  const int lane = t & 31;
  const int wave = t >> 5;
  const int Lh = L >> 3;                          // rows per head in torch-view scramble

  __shared__ float q0s[256];
  __shared__ float us[8][256];                    // u[g][k] for current head
  __shared__ float cb[8];
  __shared__ float sc[256][8];                    // scores then alpha (L<=256)
  __shared__ float zs[8][256];
  __shared__ float Sh[8];
  __shared__ float rs1[8], rs2[8];

  q0s[t] = q0g[(size_t)b * 256 + t];
  __syncthreads();

  const float* hb = hidden + (size_t)b * L * 256;

  for (int h = 0; h < 8; ++h) {
    // u[g][t] = sum_j q0[h*32+j] * W1[g*32+j, t]
#pragma unroll
    for (int g = 0; g < 8; ++g) {
      float acc = 0.f;
      const float* w1p = W1 + (size_t)(g * 32) * 256 + t;
      for (int j = 0; j < 32; ++j) acc += q0s[h * 32 + j] * w1p[(size_t)j * 256];
      us[g][t] = acc;
    }
    if (t < 8) {
      float acc = 0.f;
      for (int j = 0; j < 32; ++j) acc += q0s[h * 32 + j] * b1[t * 32 + j];
      cb[t] = acc;
    }
    __syncthreads();
    // score(l2 = rr*8+g, h) = <hidden[b, h*Lh+rr], u[g]> + cb[g]
    for (int rr = wave; rr < Lh; rr += 8) {
      const float* hp = hb + (size_t)(h * Lh + rr) * 256;
      float hv[8];
#pragma unroll
      for (int i = 0; i < 8; ++i) hv[i] = hp[lane + 32 * i];
#pragma unroll
      for (int g = 0; g < 8; ++g) {
        float p = 0.f;
#pragma unroll
        for (int i = 0; i < 8; ++i) p += hv[i] * us[g][lane + 32 * i];
#pragma unroll
        for (int off = 16; off > 0; off >>= 1) p += __shfl_xor(p, off, 32);
        if (lane == 0) {
          float x = p + cb[g];
          sc[rr * 8 + g][h] = 2.f / (1.f + __expf(-x));   // 2*sigmoid
        }
      }
    }
    __syncthreads();
  }

  // softmax over l per head (wave == head)
  {
    const int h = wave;
    float mx = -1e30f;
    for (int l = lane; l < L; l += 32) mx = fmaxf(mx, sc[l][h]);
#pragma unroll
    for (int off = 16; off > 0; off >>= 1) mx = fmaxf(mx, __shfl_xor(mx, off, 32));
    float sum = 0.f;
    for (int l = lane; l < L; l += 32) { float e = __expf(sc[l][h] - mx); sc[l][h] = e; sum += e; }
#pragma unroll
    for (int off = 16; off > 0; off >>= 1) sum += __shfl_xor(sum, off, 32);
    const float inv = 1.f / sum;
    for (int l = lane; l < L; l += 32) {
      float a = sc[l][h] * inv;
      sc[l][h] = a;
      alpha_out[((size_t)b * L + l) * 8 + h] = a;
    }
  }
  __syncthreads();

  if (t < 8) {
    float s = 0.f;
    for (int l = 0; l < L; ++l) s += sc[l][t] * (float)mask[(size_t)b * L + l];
    Sh[t] = s;
  }
  {
    float z[8] = {0.f, 0.f, 0.f, 0.f, 0.f, 0.f, 0.f, 0.f};
    for (int l = 0; l < L; ++l) {
      float hm = hb[(size_t)l * 256 + t] * (float)mask[(size_t)b * L + l];
#pragma unroll
      for (int h = 0; h < 8; ++h) z[h] += sc[l][h] * hm;
    }
#pragma unroll
    for (int h = 0; h < 8; ++h) zs[h][t] = z[h];
  }
  __syncthreads();

  // a[d] = W2[d,:]·z_{h(d)} + b2[d]*S_{h(d)}, then LayerNorm
  const int hd = t >> 5;
  float av = b2[t] * Sh[hd];
  {
    const float* w2p = W2 + (size_t)t * 256;
    for (int k = 0; k < 256; ++k) av += w2p[k] * zs[hd][k];
  }
  float s1 = av, s2 = av * av;
#pragma unroll
  for (int off = 16; off > 0; off >>= 1) { s1 += __shfl_xor(s1, off, 32); s2 += __shfl_xor(s2, off, 32); }
  if (lane == 0) { rs1[wave] = s1; rs2[wave] = s2; }
  __syncthreads();
  if (t == 0) {
    float a1 = 0.f, a2 = 0.f;
    for (int i = 0; i < 8; ++i) { a1 += rs1[i]; a2 += rs2[i]; }
    rs1[0] = a1 * (1.f / 256.f);
    rs2[0] = a2 * (1.f / 256.f);
  }
  __syncthreads();
  const float mu = rs1[0];
  const float var = rs2[0] - mu * mu;
  out[(size_t)b * 512 + t] = ln_g[t] * (av - mu) * rsqrtf(var + EPSF) + ln_b[t];
}

// Dynamic routing + capsule attention; one block per batch element. Writes out[:, 256:512].
__global__ __launch_bounds__(256) void routing_kernel(
    const float* __restrict__ uhat_g,   // (B*3, 2048): row b*3+i, col j*256+d
    const float* __restrict__ Wa, const float* __restrict__ ba,
    float* __restrict__ out)
{
  const int b = blockIdx.x;
  const int t = threadIdx.x;
  const int lane = t & 31;
  const int j = t >> 5;                 // wave == capsule

  __shared__ float uh[3][8][256];
  __shared__ float vs[8][256];
  __shared__ float sj[8][256];
  __shared__ float bij[3][8];
  __shared__ float cij[3][8];
  __shared__ float aw[8];

  for (int i = 0; i < 3; ++i) {
    const float* src = uhat_g + (size_t)(b * 3 + i) * 2048;
    for (int x = t; x < 2048; x += 256) uh[i][x >> 8][x & 255] = src[x];
  }
  if (t < 24) bij[t >> 3][t & 7] = 0.f;
  __syncthreads();

  for (int it = 0; it < 3; ++it) {
    if (t < 3) {                        // c = softmax_j(bij[i,:])
      float mx = -1e30f;
      for (int jj = 0; jj < 8; ++jj) mx = fmaxf(mx, bij[t][jj]);
      float s = 0.f, e[8];
      for (int jj = 0; jj < 8; ++jj) { e[jj] = __expf(bij[t][jj] - mx); s += e[jj]; }
      float inv = 1.f / s;
      for (int jj = 0; jj < 8; ++jj) cij[t][jj] = e[jj] * inv;
    }
    __syncthreads();
    float n2 = 0.f;
    const float c0 = cij[0][j], c1 = cij[1][j], c2 = cij[2][j];
    for (int d = lane; d < 256; d += 32) {
      float s = c0 * uh[0][j][d] + c1 * uh[1][j][d] + c2 * uh[2][j][d];
      sj[j][d] = s;
      n2 += s * s;
    }
#pragma unroll
    for (int off = 16; off > 0; off >>= 1) n2 += __shfl_xor(n2, off, 32);
    const float f = n2 / (1.f + n2) * rsqrtf(n2 + EPSF);   // squash factor
    for (int d = lane; d < 256; d += 32) vs[j][d] = f * sj[j][d];
    if (it < 2) {
      for (int i = 0; i < 3; ++i) {
        float p = 0.f;
        for (int d = lane; d < 256; d += 32) p += uh[i][j][d] * vs[j][d];
#pragma unroll
        for (int off = 16; off > 0; off >>= 1) p += __shfl_xor(p, off, 32);
        if (lane == 0) bij[i][j] += p;
      }
    }
    __syncthreads();
  }

  {
    float p = 0.f;
    for (int d = lane; d < 256; d += 32) p += vs[j][d] * Wa[d];
#pragma unroll
    for (int off = 16; off > 0; off >>= 1) p += __shfl_xor(p, off, 32);
    if (lane == 0) aw[j] = p + ba[0];
  }
  __syncthreads();
  if (t == 0) {
    float mx = -1e30f;
    for (int jj = 0; jj < 8; ++jj) mx = fmaxf(mx, aw[jj]);
    float s = 0.f;
    for (int jj = 0; jj < 8; ++jj) { aw[jj] = __expf(aw[jj] - mx); s += aw[jj]; }
    float inv = 1.f / s;
    for (int jj = 0; jj < 8; ++jj) aw[jj] *= inv;
  }
  __syncthreads();
  float fsum = 0.f;
#pragma unroll
  for (int jj = 0; jj < 8; ++jj) fsum += aw[jj] * vs[jj][t];
  out[(size_t)b * 512 + 256 + t] = fsum;
}

extern "C" void kernel_launch(void* const* d_in, const int* in_sizes, int n_in,
                              void* d_out, int out_size, void* d_ws, size_t ws_size,
                              hipStream_t stream) {
  (void)n_in; (void)out_size; (void)ws_size;
  const float* ht1    = (const float*)d_in[0];
  const float* hidden = (const float*)d_in[1];
  const float* hts    = (const float*)d_in[2];
  const int*   mask   = (const int*)d_in[3];
  const float* Wz = (const float*)d_in[4];
  const float* bz = (const float*)d_in[5];
  const float* W1 = (const float*)d_in[6];
  const float* b1 = (const float*)d_in[7];
  const float* W2 = (const float*)d_in[8];
  const float* b2 = (const float*)d_in[9];
  const float* ln_g = (const float*)d_in[10];
  const float* ln_b = (const float*)d_in[11];
  const float* Wq = (const float*)d_in[12];
  const float* bq = (const float*)d_in[13];
  const float* Wt = (const float*)d_in[14];
  const float* bt = (const float*)d_in[15];
  const float* Wa = (const float*)d_in[16];
  const float* ba = (const float*)d_in[17];

  const int D = 256;
  const int B = in_sizes[0] / D;           // 1024
  const int L = in_sizes[1] / (B * D);     // 200

  float* q0   = (float*)d_ws;              // (B, 256)
  float* qk   = q0 + (size_t)B * D;        // (B*3, 256)
  float* uhat = qk + (size_t)B * 3 * D;    // (B*3, 2048)

  float* out = (float*)d_out;              // (B, 512)
  float* alpha_out = out + (size_t)B * 2 * D;  // (B, L, 8)

  // q0 = ht1 @ Wz^T + bz
  gemm_nt_wmma<<<dim3(B / 32, D / 64), 256, 0, stream>>>(ht1, Wz, bz, q0, B, D, D);
  // q = hts @ Wq^T + bq
  gemm_nt_wmma<<<dim3((3 * B) / 32, D / 64), 256, 0, stream>>>(hts, Wq, bq, qk, 3 * B, D, D);
  // u_hat = q @ Wt^T + bt
  gemm_nt_wmma<<<dim3((3 * B) / 32, (8 * D) / 64), 256, 0, stream>>>(qk, Wt, bt, uhat, 3 * B, 8 * D, D);
  // fused attention branch -> out[:, :256], alpha
  attn_fused<<<B, 256, 0, stream>>>(q0, hidden, mask, W1, b1, W2, b2, ln_g, ln_b, out, alpha_out, L);
  // capsule routing -> out[:, 256:]
  routing_kernel<<<B, 256, 0, stream>>>(uhat, Wa, ba, out);
}